// PhysicsRenderer_17592186044907
// MI455X (gfx1250) — compile-verified
//
#include <hip/hip_runtime.h>
#include <math.h>

typedef __attribute__((ext_vector_type(2))) float v2f;
typedef __attribute__((ext_vector_type(8))) float v8f;

#define B_    8
#define N_    1024
#define T_    512
#define K_    9
#define H_    32
#define KPAD  16
#define CHUNK 32                // K-rows staged per async chunk
#define NCHUNK (N_ / CHUNK)     // 32 chunks
#define KSTEPS (CHUNK / 4)      // 8 WMMA K-steps per chunk

#if __has_builtin(__builtin_amdgcn_global_load_async_to_lds_b128)
#define HAVE_ASYNC_LDS 1
typedef __attribute__((__vector_size__(4 * sizeof(int)))) int v4i;
typedef __attribute__((address_space(1))) v4i as1_v4i;
typedef __attribute__((address_space(3))) v4i as3_v4i;
#else
#define HAVE_ASYNC_LDS 0
#endif

__device__ __forceinline__ void wait_async_4() {
#if __has_builtin(__builtin_amdgcn_s_wait_asynccnt)
    __builtin_amdgcn_s_wait_asynccnt(4);
#else
    asm volatile("s_wait_asynccnt 0x4" ::: "memory");
#endif
    asm volatile("" ::: "memory");
}
__device__ __forceinline__ void wait_async_0() {
#if __has_builtin(__builtin_amdgcn_s_wait_asynccnt)
    __builtin_amdgcn_s_wait_asynccnt(0);
#else
    asm volatile("s_wait_asynccnt 0x0" ::: "memory");
#endif
    asm volatile("" ::: "memory");
}

__constant__ float c_elec[K_][3] = {
    { 0.30f,  0.25f, 0.00f},
    {-0.30f,  0.25f, 0.00f},
    {-0.05f, -0.40f, 0.00f},
    { 0.03f,  0.05f, 0.20f},
    { 0.00f,  0.05f, 0.22f},
    {-0.04f,  0.01f, 0.21f},
    {-0.08f, -0.02f, 0.18f},
    {-0.14f, -0.02f, 0.13f},
    {-0.20f, -0.02f, 0.07f}};

// ---------------------------------------------------------------------------
// Kernel 1: G[b][k][n] = corr(b,n,k) * dot3(r_vec, dir) * amp / (r2+0.04)^1.5
// Layout [b][KPAD][n], rows k>=9 zeroed (WMMA A-matrix padding).
// ---------------------------------------------------------------------------
__global__ void gains_kernel(const float* __restrict__ pos,
                             const float* __restrict__ dir,
                             const float* __restrict__ amp,
                             const float* __restrict__ eoff,
                             const float* __restrict__ W1,
                             const float* __restrict__ b1,
                             const float* __restrict__ W2,
                             const float* __restrict__ b2,
                             float* __restrict__ G) {
    int tid = blockIdx.x * blockDim.x + threadIdx.x;
    if (tid >= B_ * KPAD * N_) return;
    int n = tid % N_;
    int k = (tid / N_) % KPAD;
    int b = tid / (N_ * KPAD);
    if (k >= K_) { G[tid] = 0.0f; return; }

    const float px = pos[(b * N_ + n) * 3 + 0];
    const float py = pos[(b * N_ + n) * 3 + 1];
    const float pz = pos[(b * N_ + n) * 3 + 2];

    const float ex = c_elec[k][0] + eoff[(b * K_ + k) * 3 + 0];
    const float ey = c_elec[k][1] + eoff[(b * K_ + k) * 3 + 1];
    const float ez = c_elec[k][2] + eoff[(b * K_ + k) * 3 + 2];

    const float rx = ex - px, ry = ey - py, rz = ez - pz;
    const float r2 = rx * rx + ry * ry + rz * rz;

    const float q = r2 + 0.04f;
    const float denom = q * sqrtf(q);

    float rd = sqrtf(r2);
    rd = fmaxf(rd, 1e-6f);
    const float ird = 1.0f / rd;
    const float hx = rx * ird, hy = ry * ird, hz = rz * ird;

    float raw = b2[0];
    #pragma unroll 4
    for (int h = 0; h < H_; ++h) {
        const float* w = W1 + h * 7;
        float a = w[0] * px + w[1] * py + w[2] * pz
                + w[3] * hx + w[4] * hy + w[5] * hz
                + w[6] * rd + b1[h];
        float g = 0.5f * a * (1.0f + erff(a * 0.70710678118654752f));
        raw += W2[h] * g;
    }
    const float corr = 1.0f + 0.5f * tanhf(raw);

    const float dx = dir[(b * N_ + n) * 3 + 0];
    const float dy = dir[(b * N_ + n) * 3 + 1];
    const float dz = dir[(b * N_ + n) * 3 + 2];
    const float dot3 = rx * dx + ry * dy + rz * dz;

    G[tid] = (corr / denom) * dot3 * amp[b * N_ + n];
}

// ---------------------------------------------------------------------------
// Kernel 2: batched GEMM V_e[b,0:16,t0:t0+16] = G[b] (16x1024) @ env[b] (1024x512)
// One wave32 per 16x16 tile; V_WMMA_F32_16X16X4_F32.
// Envelope stream staged LDS-double-buffered via global_load_async_to_lds_b128
// (2 KB chunks = 32 K-rows x 16 t-cols), synchronized with s_wait_asynccnt.
// Per chunk: batch-issue 8 A loads + 8 ds loads, then 8 back-to-back WMMAs.
// ---------------------------------------------------------------------------
__global__ __launch_bounds__(32) void gemm_kernel(const float* __restrict__ G,
                                                  const float* __restrict__ env,
                                                  float* __restrict__ Ve) {
    const int wave = blockIdx.x;            // B_ * (T_/16) waves
    const int b    = wave / (T_ / 16);
    const int t0   = (wave % (T_ / 16)) * 16;
    const int lane = threadIdx.x;
    const int row  = lane & 15;             // M row (A) / N column (B,C,D)
    const int kh   = lane >> 4;             // K half-pair select

    // A operand: float2 from G row `row`, columns n + 2*kh + {0,1}
    const float* ga = G + ((size_t)(b * KPAD + row)) * N_ + 2 * kh;
    const float* envb = env + (size_t)b * N_ * T_ + t0;

    v8f acc = {};

#if HAVE_ASYNC_LDS
    __shared__ __align__(16) float sbuf[2][CHUNK][16];   // 2 x 2 KB double buffer

    // Issue async copy of chunk cc into buffer pb: 32 rows x 64 B, 4 x B128/lane.
    auto issue = [&](int cc, int pb) {
        const float* gbase = envb + (size_t)(cc * CHUNK) * T_;
        #pragma unroll
        for (int i = 0; i < 4; ++i) {
            int linear = i * 32 + lane;
            int r   = linear >> 2;          // 0..31
            int seg = (linear & 3) * 4;     // 0,4,8,12 floats
            __builtin_amdgcn_global_load_async_to_lds_b128(
                (as1_v4i*)(gbase + (size_t)r * T_ + seg),
                (as3_v4i*)&sbuf[pb][r][seg], 0, 0);
        }
    };

    auto compute = [&](int cc, int pb) {
        const int base = cc * CHUNK;
        // Phase 1: batch all A fragment loads (global, L2-resident G).
        v2f a[KSTEPS];
        #pragma unroll
        for (int j = 0; j < KSTEPS; ++j) {
            a[j].x = ga[base + 4 * j];
            a[j].y = ga[base + 4 * j + 1];
        }
        // Phase 2: batch all B fragment loads (LDS, conflict-free).
        v2f bv[KSTEPS];
        #pragma unroll
        for (int j = 0; j < KSTEPS; ++j) {
            const int r0 = 4 * j + 2 * kh;
            bv[j].x = sbuf[pb][r0][row];
            bv[j].y = sbuf[pb][r0 + 1][row];
        }
        // Phase 3: back-to-back WMMAs (D->C accumulate chain, no A/B hazard).
        #pragma unroll
        for (int j = 0; j < KSTEPS; ++j) {
            acc = __builtin_amdgcn_wmma_f32_16x16x4_f32(
                false, a[j], false, bv[j], (short)0, acc, false, false);
        }
    };

    issue(0, 0);
    issue(1, 1);
    for (int c = 0; c < NCHUNK - 2; ++c) {
        wait_async_4();                     // chunk c landed
        compute(c, c & 1);
        issue(c + 2, c & 1);                // prefetch 2 chunks ahead
    }
    wait_async_4();
    compute(NCHUNK - 2, (NCHUNK - 2) & 1);
    wait_async_0();
    compute(NCHUNK - 1, (NCHUNK - 1) & 1);
#else
    // Fallback: direct global loads in WMMA B layout.
    const float* eb = envb + (size_t)(2 * kh) * T_ + row;
    #pragma unroll 4
    for (int n0 = 0; n0 < N_; n0 += 4) {
        v2f a;
        a.x = ga[n0];
        a.y = ga[n0 + 1];
        v2f bv;
        bv.x = eb[(size_t)n0 * T_];
        bv.y = eb[(size_t)(n0 + 1) * T_];
        acc = __builtin_amdgcn_wmma_f32_16x16x4_f32(
            false, a, false, bv, (short)0, acc, false, false);
    }
#endif

    // D layout: VGPR v, lanes 0-15 -> M=v, lanes 16-31 -> M=v+8; N = row
    float* vb = Ve + ((size_t)b * K_) * T_ + t0 + row;
    #pragma unroll
    for (int v = 0; v < 8; ++v) {
        int m = v + 8 * kh;
        if (m < K_) {
            float val = fminf(fmaxf(acc[v], -50.0f), 50.0f);
            vb[(size_t)m * T_] = val;
        }
    }
}

// ---------------------------------------------------------------------------
// Kernel 3: 12-lead recombination from clipped V_electrodes.
// ---------------------------------------------------------------------------
__global__ void leads_kernel(const float* __restrict__ Ve,
                             float* __restrict__ out) {
    int idx = blockIdx.x * blockDim.x + threadIdx.x;
    if (idx >= B_ * T_) return;
    int b = idx / T_;
    int t = idx % T_;

    const float* v = Ve + (size_t)b * K_ * T_ + t;
    const float RA = v[0 * T_];
    const float LA = v[1 * T_];
    const float LL = v[2 * T_];
    const float WCT = (RA + LA + LL) * (1.0f / 3.0f);

    float* o = out + (size_t)b * 12 * T_ + t;
    o[0 * T_] = LA - RA;
    o[1 * T_] = LL - RA;
    o[2 * T_] = LL - LA;
    o[3 * T_] = RA - 0.5f * (LA + LL);
    o[4 * T_] = LA - 0.5f * (RA + LL);
    o[5 * T_] = LL - 0.5f * (RA + LA);
    #pragma unroll
    for (int cix = 0; cix < 6; ++cix)
        o[(6 + cix) * T_] = v[(3 + cix) * T_] - WCT;
}

extern "C" void kernel_launch(void* const* d_in, const int* in_sizes, int n_in,
                              void* d_out, int out_size, void* d_ws, size_t ws_size,
                              hipStream_t stream) {
    const float* pos  = (const float*)d_in[0];  // (B,N,3)
    const float* dir  = (const float*)d_in[1];  // (B,N,3)
    const float* amp  = (const float*)d_in[2];  // (B,N)
    const float* env  = (const float*)d_in[3];  // (B,N,T)
    const float* eoff = (const float*)d_in[4];  // (B,K,3)
    const float* W1   = (const float*)d_in[5];  // (H,7)
    const float* b1   = (const float*)d_in[6];  // (H,)
    const float* W2   = (const float*)d_in[7];  // (1,H)
    const float* b2   = (const float*)d_in[8];  // (1,)

    float* out = (float*)d_out;
    float* v_leads = out;                         // (B,12,T)
    float* Ve      = out + (size_t)B_ * 12 * T_;  // (B,9,T)

    float* G = (float*)d_ws;                      // (B,16,N) = 512 KB

    {
        int total = B_ * KPAD * N_;
        int block = 256;
        int grid = (total + block - 1) / block;
        gains_kernel<<<grid, block, 0, stream>>>(pos, dir, amp, eoff, W1, b1, W2, b2, G);
    }
    {
        int grid = B_ * (T_ / 16);   // 256 waves
        gemm_kernel<<<grid, 32, 0, stream>>>(G, env, Ve);
    }
    {
        int total = B_ * T_;
        int block = 256;
        int grid = (total + block - 1) / block;
        leads_kernel<<<grid, block, 0, stream>>>(Ve, v_leads);
    }
}